// GCN_22866405884174
// MI455X (gfx1250) — compile-verified
//
#include <hip/hip_runtime.h>
#include <hip/hip_bf16.h>

typedef __attribute__((ext_vector_type(2))) float v2f;
typedef __attribute__((ext_vector_type(8))) float v8f;

#define F_IN 128
#define HID  64
#define NOUT 10

// D = A(16x4 f32) * B(4x16 f32) + C(16x16 f32), full fp32 precision on the matrix pipe.
static __device__ __forceinline__ v8f wmma_f32_k4(v2f a, v2f b, v8f c) {
    return __builtin_amdgcn_wmma_f32_16x16x4_f32(
        /*neg_a=*/false, a, /*neg_b=*/false, b,
        /*c_mod=*/(short)0, c, /*reuse_a=*/false, /*reuse_b=*/false);
}

// ---- normalization ----------------------------------------------------------

__global__ void k_deg_init(float* __restrict__ deg, int n) {
    int i = blockIdx.x * blockDim.x + threadIdx.x;
    if (i < n) deg[i] = 1.0f;               // self-loop contributes 1 to degree
}

__global__ void k_deg_acc(const int* __restrict__ dst, float* __restrict__ deg, int e) {
    int i = blockIdx.x * blockDim.x + threadIdx.x;
    if (i < e) atomicAdd(&deg[dst[i]], 1.0f);
}

__global__ void k_dinv(float* __restrict__ deg, int n) {
    int i = blockIdx.x * blockDim.x + threadIdx.x;
    if (i < n) deg[i] = rsqrtf(deg[i]);     // deg >= 1 always (self-loops)
}

// ---- GEMM1: h = x @ W1  (N x 128 @ 128 x 64), one 16x16 tile per wave -------

__global__ void __launch_bounds__(256)
k_gemm1(const float* __restrict__ x, const float* __restrict__ W1,
        float* __restrict__ h, int totalWaves) {
    int gtid = blockIdx.x * blockDim.x + threadIdx.x;
    int wave = gtid >> 5;
    if (wave >= totalWaves) return;          // wave-uniform; EXEC stays all-1s
    int lane = threadIdx.x & 31;

    int rowTile = wave >> 2;                 // HID/16 == 4 column tiles
    int col0    = (wave & 3) << 4;
    int row0    = rowTile << 4;

    int m  = lane & 15;                      // M (A rows) == N (B cols) per-lane index
    int kb = (lane >> 4) << 1;               // K sub-pair: lanes 0-15 -> K{0,1}, 16-31 -> K{2,3}

    const float* xrow = x  + (size_t)(row0 + m) * F_IN + kb;
    const float* wcol = W1 + (size_t)kb * HID + col0 + m;

    v8f c = {};
    #pragma unroll 8
    for (int k0 = 0; k0 < F_IN; k0 += 4) {
        v2f a, b;
        a.x = xrow[k0];
        a.y = xrow[k0 + 1];
        b.x = wcol[(size_t)k0 * HID];
        b.y = wcol[(size_t)(k0 + 1) * HID];
        c = wmma_f32_k4(a, b, c);
    }

    int mb = (lane >> 4) << 3;               // lanes 16-31 hold rows M+8..M+15
    float* hout = h + (size_t)(row0 + mb) * HID + col0 + m;
    #pragma unroll
    for (int r = 0; r < 8; ++r) hout[(size_t)r * HID] = c[r];
}

// ---- self-loop message: agg = dinv[i]^2 * h[i]  (also serves as agg init) ---

__global__ void k_selfmsg(const float* __restrict__ h, const float* __restrict__ dinv,
                          float* __restrict__ agg, int total) {
    int idx = blockIdx.x * blockDim.x + threadIdx.x;
    if (idx < total) {
        int i = idx >> 6;                    // / HID
        float w = dinv[i];
        agg[idx] = w * w * h[idx];
    }
}

// ---- scatter layer 1: 32 lanes per edge, float2 per lane (coalesced 256B) ---

__global__ void k_scatter1(const int* __restrict__ src, const int* __restrict__ dst,
                           const float* __restrict__ dinv, const float* __restrict__ h,
                           float* __restrict__ agg, int e) {
    int gtid = blockIdx.x * blockDim.x + threadIdx.x;
    int edge = gtid >> 5;
    if (edge >= e) return;
    int lane = threadIdx.x & 31;
    int s = src[edge], d = dst[edge];
    float w = dinv[s] * dinv[d];
    const v2f* hv = (const v2f*)(h + (size_t)s * HID);
    v2f v = hv[lane];
    float* a = agg + (size_t)d * HID + (lane << 1);
    atomicAdd(a,     w * v.x);
    atomicAdd(a + 1, w * v.y);
}

// ---- bias + ReLU into hr (reuses h buffer) ----------------------------------

__global__ void k_relu_bias(const float* __restrict__ agg, const float* __restrict__ b1,
                            float* __restrict__ hr, int total) {
    int idx = blockIdx.x * blockDim.x + threadIdx.x;
    if (idx < total) hr[idx] = fmaxf(agg[idx] + b1[idx & (HID - 1)], 0.0f);
}

// ---- GEMM2: h2 = hr @ W2  (N x 64 @ 64 x 10, N-cols padded to 16) -----------

__global__ void __launch_bounds__(256)
k_gemm2(const float* __restrict__ hr, const float* __restrict__ W2,
        float* __restrict__ h2, int totalWaves) {
    int gtid = blockIdx.x * blockDim.x + threadIdx.x;
    int wave = gtid >> 5;
    if (wave >= totalWaves) return;
    int lane = threadIdx.x & 31;

    int row0 = wave << 4;
    int m    = lane & 15;
    int kb   = (lane >> 4) << 1;
    bool valid_col = (m < NOUT);

    const float* hrow = hr + (size_t)(row0 + m) * HID + kb;

    v8f c = {};
    #pragma unroll
    for (int k0 = 0; k0 < HID; k0 += 4) {
        v2f a, b;
        a.x = hrow[k0];
        a.y = hrow[k0 + 1];
        b.x = valid_col ? W2[(size_t)(k0 + kb) * NOUT + m]     : 0.0f;  // pad cols 10..15
        b.y = valid_col ? W2[(size_t)(k0 + kb + 1) * NOUT + m] : 0.0f;
        c = wmma_f32_k4(a, b, c);
    }

    int mb = (lane >> 4) << 3;
    if (valid_col) {
        #pragma unroll
        for (int r = 0; r < 8; ++r)
            h2[(size_t)(row0 + mb + r) * NOUT + m] = c[r];
    }
}

// ---- output init: out = b2 + dinv^2 * h2 (fully overwrites poisoned d_out) --

__global__ void k_out_init(const float* __restrict__ h2, const float* __restrict__ dinv,
                           const float* __restrict__ b2, float* __restrict__ out, int total) {
    int idx = blockIdx.x * blockDim.x + threadIdx.x;
    if (idx < total) {
        int i = idx / NOUT;
        int j = idx - i * NOUT;
        float w = dinv[i];
        out[idx] = b2[j] + w * w * h2[idx];
    }
}

// ---- scatter layer 2: 16 lanes per edge, 10 active --------------------------

__global__ void k_scatter2(const int* __restrict__ src, const int* __restrict__ dst,
                           const float* __restrict__ dinv, const float* __restrict__ h2,
                           float* __restrict__ out, int e) {
    int gtid = blockIdx.x * blockDim.x + threadIdx.x;
    int edge = gtid >> 4;
    if (edge >= e) return;
    int j = threadIdx.x & 15;
    if (j >= NOUT) return;
    int s = src[edge], d = dst[edge];
    float w = dinv[s] * dinv[d];
    atomicAdd(&out[(size_t)d * NOUT + j], w * h2[(size_t)s * NOUT + j]);
}

// ---- host-side launch sequence ----------------------------------------------

extern "C" void kernel_launch(void* const* d_in, const int* in_sizes, int n_in,
                              void* d_out, int out_size, void* d_ws, size_t ws_size,
                              hipStream_t stream) {
    const float* x  = (const float*)d_in[0];
    const int*   ei = (const int*)  d_in[1];
    const float* W1 = (const float*)d_in[2];
    const float* b1 = (const float*)d_in[3];
    const float* W2 = (const float*)d_in[4];
    const float* b2 = (const float*)d_in[5];

    const int n = in_sizes[0] / F_IN;        // 100000 (divisible by 16)
    const int e = in_sizes[1] / 2;           // 1600000
    const int* src = ei;                     // edge_index[0]
    const int* dst = ei + e;                 // edge_index[1]

    float* ws   = (float*)d_ws;
    float* dinv = ws;                        // N
    float* h    = dinv + n;                  // N*64  (x@W1, later relu(h))
    float* agg  = h + (size_t)n * HID;       // N*64
    float* h2   = agg + (size_t)n * HID;     // N*10
    float* out  = (float*)d_out;

    const int B = 256;
    const int rowTiles = n >> 4;             // N/16 == 6250 exactly

    // normalization
    k_deg_init<<<(n + B - 1) / B, B, 0, stream>>>(dinv, n);
    k_deg_acc <<<(e + B - 1) / B, B, 0, stream>>>(dst, dinv, e);
    k_dinv    <<<(n + B - 1) / B, B, 0, stream>>>(dinv, n);

    // layer 1
    const int waves1 = rowTiles * (HID / 16);
    k_gemm1<<<(waves1 * 32 + B - 1) / B, B, 0, stream>>>(x, W1, h, waves1);

    const int tot1 = n * HID;
    k_selfmsg <<<(tot1 + B - 1) / B, B, 0, stream>>>(h, dinv, agg, tot1);
    k_scatter1<<<(int)(((size_t)e * 32 + B - 1) / B), B, 0, stream>>>(src, dst, dinv, h, agg, e);
    k_relu_bias<<<(tot1 + B - 1) / B, B, 0, stream>>>(agg, b1, h, tot1);

    // layer 2
    k_gemm2<<<(rowTiles * 32 + B - 1) / B, B, 0, stream>>>(h, W2, h2, rowTiles);

    const int tot2 = n * NOUT;
    k_out_init<<<(tot2 + B - 1) / B, B, 0, stream>>>(h2, dinv, b2, out, tot2);
    k_scatter2<<<(int)(((size_t)e * 16 + B - 1) / B), B, 0, stream>>>(src, dst, dinv, h2, out, e);
}